// FirstLayer_64725157151121
// MI455X (gfx1250) — compile-verified
//
#include <hip/hip_runtime.h>

// Problem constants (from reference).
#define N_VUL  100000
#define N_EDGE 500000
#define DIM    128

// CDNA5 async global->LDS path (ASYNCcnt). Gated: fall back to direct
// global_load_b128 + global_prefetch_b8 if the builtin is absent.
#if defined(__HIP_DEVICE_COMPILE__) && \
    __has_builtin(__builtin_amdgcn_global_load_async_to_lds_b128)
#define USE_ASYNC_LDS 1
#else
#define USE_ASYNC_LDS 0
#endif

// ---------------------------------------------------------------------------
// Native f32 atomic add (relaxed, agent scope) -> GLOBAL_ATOMIC_ADD_F32
// (no-return form, STOREcnt path). scatter_mean is first in .text so the
// disasm snippet shows the atomic lowering.
// ---------------------------------------------------------------------------
__device__ __forceinline__ void atomAddF(float* p, float v) {
  (void)__hip_atomic_fetch_add(p, v, __ATOMIC_RELAXED, __HIP_MEMORY_SCOPE_AGENT);
}

#if USE_ASYNC_LDS
// Builtin signature (from clang diagnostic): first param is a pointer to
// int __vector_size__(16) in the global (as1) address space; LDS side is as3.
typedef int v4i __attribute__((vector_size(16)));
typedef __attribute__((address_space(1))) v4i* gvec4_ptr;
typedef __attribute__((address_space(3))) v4i* lvec4_ptr;

// Each lane copies its own 16B chunk: global (as1) -> LDS (as3), ASYNCcnt.
__device__ __forceinline__ void async_gather16(const float* g, float* l) {
  __builtin_amdgcn_global_load_async_to_lds_b128(
      (gvec4_ptr)g, (lvec4_ptr)l, /*offset=*/0, /*cpol=*/0);
}
__device__ __forceinline__ void wait_async_le1() {
#if __has_builtin(__builtin_amdgcn_s_wait_asynccnt)
  __builtin_amdgcn_s_wait_asynccnt(1);
#else
  asm volatile("s_wait_asynccnt 0x1" ::: "memory");
#endif
}
__device__ __forceinline__ void wait_async_le0() {
#if __has_builtin(__builtin_amdgcn_s_wait_asynccnt)
  __builtin_amdgcn_s_wait_asynccnt(0);
#else
  asm volatile("s_wait_asynccnt 0x0" ::: "memory");
#endif
}
#endif

// ---------------------------------------------------------------------------
// Fused scatter-mean: one wave32 per edge, lane l owns columns [4l, 4l+4).
//  - 512B row gather per edge, fully coalesced
//  - scale = 1/(3*max(deg,1)) folded in: out accumulates the final answer
//  - 4 contiguous global_atomic_add_f32 per lane (wave covers the row
//    contiguously -> dense, mergeable dword streams at the L2 atomic units)
//  - ASYNC pipeline: while processing edge e from LDS slot s, the gather for
//    edge e+nwaves is already in flight into slot s^1 via
//    global_load_async_to_lds_b128; s_wait_asynccnt<=1 retires the older one.
// ---------------------------------------------------------------------------
__global__ void __launch_bounds__(256)
scatter_mean(const float* __restrict__ x, const int* __restrict__ src,
             const int* __restrict__ dst, const int* __restrict__ cnt,
             float* __restrict__ out, int nedge) {
  const int lane   = threadIdx.x & 31;
  const int wid    = threadIdx.x >> 5;                       // wave in block
  const int wave   = (blockIdx.x * blockDim.x + threadIdx.x) >> 5;
  const int nwaves = (gridDim.x * blockDim.x) >> 5;
  const int col    = lane * 4;

#if USE_ASYNC_LDS
  __shared__ float stage[8][2][DIM];    // 8 waves x double buffer x 512B = 8KB

  // Prologue: start the gather for this wave's first edge into slot 0.
  if (wave < nedge) {
    const int s0 = src[wave];
    async_gather16(&x[(size_t)s0 * DIM + col], &stage[wid][0][col]);
  }

  int slot = 0;
  for (int e = wave; e < nedge; e += nwaves) {
    const int en = e + nwaves;
    if (en < nedge) {
      const int sn = src[en];
      async_gather16(&x[(size_t)sn * DIM + col], &stage[wid][slot ^ 1][col]);
      wait_async_le1();                 // oldest (current slot) has landed
    } else {
      wait_async_le0();
    }

    const int d = dst[e];
    int c = cnt[d];
    if (c < 1) c = 1;
    const float scale = 1.0f / (3.0f * (float)c);

    const float4 v = *reinterpret_cast<const float4*>(&stage[wid][slot][col]);
    float* o = &out[(size_t)d * DIM + col];
    atomAddF(o + 0, v.x * scale);
    atomAddF(o + 1, v.y * scale);
    atomAddF(o + 2, v.z * scale);
    atomAddF(o + 3, v.w * scale);
    slot ^= 1;
  }
#else
  for (int e = wave; e < nedge; e += nwaves) {
    const int en = e + nwaves;
    if (en < nedge) {
      const int sn = src[en];
      __builtin_prefetch(&x[(size_t)sn * DIM + col], 0, 1);
    }
    const int s = src[e];
    const int d = dst[e];
    int c = cnt[d];
    if (c < 1) c = 1;
    const float scale = 1.0f / (3.0f * (float)c);

    const float4 v = *reinterpret_cast<const float4*>(&x[(size_t)s * DIM + col]);
    float* o = &out[(size_t)d * DIM + col];
    atomAddF(o + 0, v.x * scale);
    atomAddF(o + 1, v.y * scale);
    atomAddF(o + 2, v.z * scale);
    atomAddF(o + 3, v.w * scale);
  }
#endif
}

// ---------------------------------------------------------------------------
// In-degree histogram for all three relations in one pass.
// cnt layout: [0,N_VUL) = rel a, [N_VUL,2N) = rel b, [2N,3N) = rel c.
// ---------------------------------------------------------------------------
__global__ void __launch_bounds__(256)
count3(const int* __restrict__ da, const int* __restrict__ db,
       const int* __restrict__ dc, int* __restrict__ cnt, int nedge) {
  int i = blockIdx.x * blockDim.x + threadIdx.x;
  const int stride = gridDim.x * blockDim.x;
  for (; i < nedge; i += stride) {
    atomicAdd(&cnt[da[i]], 1);
    atomicAdd(&cnt[N_VUL + db[i]], 1);
    atomicAdd(&cnt[2 * N_VUL + dc[i]], 1);
  }
}

// ---------------------------------------------------------------------------
// Zeroing kernels (graph-capture-safe; d_out/d_ws are poisoned by harness).
// ---------------------------------------------------------------------------
__global__ void zero_f32x4(float4* __restrict__ p, int n4) {
  int i = blockIdx.x * blockDim.x + threadIdx.x;
  const int stride = gridDim.x * blockDim.x;
  const float4 z = make_float4(0.f, 0.f, 0.f, 0.f);
  for (; i < n4; i += stride) p[i] = z;
}

__global__ void zero_i32(int* __restrict__ p, int n) {
  int i = blockIdx.x * blockDim.x + threadIdx.x;
  const int stride = gridDim.x * blockDim.x;
  for (; i < n; i += stride) p[i] = 0;
}

// ---------------------------------------------------------------------------
// Launch: zero out + counts, histogram degrees, then three fused scatters.
// Stream order serializes phases; capture-safe. Workspace: 1.2 MB.
// ---------------------------------------------------------------------------
extern "C" void kernel_launch(void* const* d_in, const int* in_sizes, int n_in,
                              void* d_out, int out_size, void* d_ws, size_t ws_size,
                              hipStream_t stream) {
  const float* xa = (const float*)d_in[0];
  const float* xb = (const float*)d_in[1];
  const float* xc = (const float*)d_in[2];
  const int*   sa = (const int*)d_in[3];
  const int*   da = (const int*)d_in[4];
  const int*   sb = (const int*)d_in[5];
  const int*   db = (const int*)d_in[6];
  const int*   sc = (const int*)d_in[7];
  const int*   dc = (const int*)d_in[8];

  float* out = (float*)d_out;
  int*   cnt = (int*)d_ws;            // 3 * N_VUL ints

  zero_f32x4<<<4096, 256, 0, stream>>>((float4*)out, N_VUL * DIM / 4);
  zero_i32 <<<512,  256, 0, stream>>>(cnt, 3 * N_VUL);

  count3<<<(N_EDGE + 255) / 256, 256, 0, stream>>>(da, db, dc, cnt, N_EDGE);

  // 15625 blocks * 8 waves = 125000 waves -> 4 edges per wave
  scatter_mean<<<15625, 256, 0, stream>>>(xa, sa, da, cnt,             out, N_EDGE);
  scatter_mean<<<15625, 256, 0, stream>>>(xb, sb, db, cnt + N_VUL,     out, N_EDGE);
  scatter_mean<<<15625, 256, 0, stream>>>(xc, sc, dc, cnt + 2 * N_VUL, out, N_EDGE);
}